// PinSAGEModule_27616639713818
// MI455X (gfx1250) — compile-verified
//
#include <hip/hip_runtime.h>
#include <cstddef>

typedef __attribute__((ext_vector_type(2))) float v2f;
typedef __attribute__((ext_vector_type(8))) float v8f;

#define SAGE_WAVES 5   // 5 waves * 16 rows = 80 rows/block; 50000 and 10000 both divide 80*... (50000/80=625)
#define DHID 128

// ---------------------------------------------------------------------------
// zero a float buffer
__global__ void zero_kernel(float* __restrict__ p, size_t n) {
    size_t i = (size_t)blockIdx.x * blockDim.x + threadIdx.x;
    if (i < n) p[i] = 0.0f;
}

// ---------------------------------------------------------------------------
// Edge aggregation: sum[dst] += x_src[src], cnt[dst] += 1.
// 32 lanes cooperate on one edge: lane handles 4 channels (float4 gather).
__global__ void agg_kernel(const float* __restrict__ xsrc,
                           const int* __restrict__ src,
                           const int* __restrict__ dst,
                           float* __restrict__ sum,
                           float* __restrict__ cnt,
                           int E) {
    size_t gid = (size_t)blockIdx.x * blockDim.x + threadIdx.x;
    int e = (int)(gid >> 5);
    int lane = (int)(gid & 31);
    if (e >= E) return;
    int d = dst[e];
    int s = src[e];
    const float4 v = *(const float4*)(xsrc + (size_t)s * DHID + lane * 4);
    float* sp = sum + (size_t)d * DHID + lane * 4;
    atomicAdd(sp + 0, v.x);
    atomicAdd(sp + 1, v.y);
    atomicAdd(sp + 2, v.z);
    atomicAdd(sp + 3, v.w);
    if (lane == 0) atomicAdd(cnt + d, 1.0f);
}

// ---------------------------------------------------------------------------
// Stage a 128x128 row-major weight W[n][k] into LDS as packed (k-pair, n):
//   lw[(k>>1)*256 + n*2 + (k&1)] = W[n*128 + k]
// so B fragment (B[k][n], B[k+1][n]) for even k is one aligned 8B read.
__device__ inline void stage_weight(const float* __restrict__ W, float* lw,
                                    int tix, int nthr) {
    for (int idx = tix; idx < DHID * DHID; idx += nthr) {
        int n = idx >> 7;
        int k = idx & 127;
        lw[((k >> 1) << 8) + (n << 1) + (k & 1)] = W[idx];
    }
}

// K-loop: acc += (Arow * rc) @ B, using V_WMMA_F32_16X16X4_F32.
// A 16x4 f32 layout: a.x = A[m][k0 + (hi?2:0)], a.y = A[m][k0 + (hi?2:0) + 1]
// B 4x16 f32 layout: b.x = B[ka][n], b.y = B[ka+1][n], n = lane&15
__device__ inline void kloop(const float* __restrict__ Arow, float rc,
                             const float* lw, int lane, v8f acc[8]) {
    const bool hi = lane >= 16;
    const int nloc = lane & 15;
    for (int k0 = 0; k0 < DHID; k0 += 4) {
        int ka = k0 + (hi ? 2 : 0);
        v2f a = *(const v2f*)(Arow + ka);
        a = a * rc;
        const float* bp = lw + ((ka >> 1) << 8);
#pragma unroll
        for (int t = 0; t < 8; ++t) {
            v2f bv = *(const v2f*)(bp + (((t << 4) + nloc) << 1));
            acc[t] = __builtin_amdgcn_wmma_f32_16x16x4_f32(
                false, a, false, bv, (short)0, acc[t], false, false);
        }
    }
}

// ---------------------------------------------------------------------------
// SAGE conv GEMM + epilogue.
// out[m] = L2normalize( (sum[m]/max(cnt,1)) @ Wl^T + b + xdst[m] @ Wr^T ), opt ReLU.
// One wave per 16-row tile; 8 n-tiles of 16 -> full 128-wide output row.
__global__ __launch_bounds__(SAGE_WAVES * 32)
void sage_kernel(const float* __restrict__ sum, const float* __restrict__ cnt,
                 const float* __restrict__ xdst,
                 const float* __restrict__ Wl, const float* __restrict__ Wr,
                 const float* __restrict__ bias,
                 float* __restrict__ out, int relu, int nrows,
                 const int* limit_ptr, int limit_const,
                 const int* off_ptr) {
    __shared__ float lw[DHID * DHID];   // 64 KB

    const int lane = threadIdx.x & 31;
    const int wid = threadIdx.x >> 5;
    const int mtile = blockIdx.x * SAGE_WAVES + wid;
    const int nloc = lane & 15;
    const bool hi = lane >= 16;

    int row = mtile * 16 + nloc;
    if (row >= nrows) row = 0;          // safe clamp; store is guarded below

    v8f acc[8] = {};

    // GEMM 1: mean-aggregated neighbors through Wl
    stage_weight(Wl, lw, threadIdx.x, blockDim.x);
    __syncthreads();
    float rc = 1.0f / fmaxf(cnt[row], 1.0f);
    kloop(sum + (size_t)row * DHID, rc, lw, lane, acc);
    __syncthreads();

    // GEMM 2: root features through Wr
    stage_weight(Wr, lw, threadIdx.x, blockDim.x);
    __syncthreads();
    kloop(xdst + (size_t)row * DHID, 1.0f, lw, lane, acc);

    // Epilogue: bias, L2 normalize per row, optional ReLU, store.
#pragma unroll
    for (int t = 0; t < 8; ++t) acc[t] = acc[t] + bias[(t << 4) + nloc];

    const int limit = limit_ptr ? *limit_ptr : limit_const;
    const size_t outoff = off_ptr ? (size_t)(*off_ptr) * DHID : 0;
    const int m_off = hi ? 8 : 0;

#pragma unroll
    for (int r = 0; r < 8; ++r) {
        float s = 0.0f;
#pragma unroll
        for (int t = 0; t < 8; ++t) { float v = acc[t][r]; s += v * v; }
        // row elements live across 16 lanes of this half-wave (wave32)
#pragma unroll
        for (int off = 1; off < 16; off <<= 1) s += __shfl_xor(s, off, 16);
        float inv = 1.0f / fmaxf(sqrtf(s), 1e-12f);
        int om = mtile * 16 + m_off + r;
        if (om < limit) {
            float* orow = out + outoff + (size_t)om * DHID;
#pragma unroll
            for (int t = 0; t < 8; ++t) {
                float o = acc[t][r] * inv;
                if (relu) o = fmaxf(o, 0.0f);
                orow[(t << 4) + nloc] = o;
            }
        }
    }
}

// ---------------------------------------------------------------------------
extern "C" void kernel_launch(void* const* d_in, const int* in_sizes, int n_in,
                              void* d_out, int out_size, void* d_ws, size_t ws_size,
                              hipStream_t stream) {
    const float* x_user  = (const float*)d_in[0];
    const float* x_item  = (const float*)d_in[1];
    const float* l1_ui_Wl = (const float*)d_in[2];
    const float* l1_ui_Wr = (const float*)d_in[3];
    const float* l1_ui_b  = (const float*)d_in[4];
    const float* l1_iu_Wl = (const float*)d_in[5];
    const float* l1_iu_Wr = (const float*)d_in[6];
    const float* l1_iu_b  = (const float*)d_in[7];
    const float* l2_ui_Wl = (const float*)d_in[8];
    const float* l2_ui_Wr = (const float*)d_in[9];
    const float* l2_ui_b  = (const float*)d_in[10];
    const float* l2_iu_Wl = (const float*)d_in[11];
    const float* l2_iu_Wr = (const float*)d_in[12];
    const float* l2_iu_b  = (const float*)d_in[13];
    const int* eui = (const int*)d_in[14];
    const int* eiu = (const int*)d_in[15];
    const int* bsu = (const int*)d_in[16];   // device scalar: batch_size_user
    const int* bsi = (const int*)d_in[17];   // device scalar: batch_size_item

    const int E_ui = in_sizes[14] / 2;
    const int E_iu = in_sizes[15] / 2;
    const int NU = in_sizes[0] / DHID;
    const int NI = in_sizes[1] / DHID;
    const int maxN = NU > NI ? NU : NI;

    float* ws = (float*)d_ws;
    float* sumb   = ws;
    float* cntb   = sumb + (size_t)maxN * DHID;
    float* h_user = cntb + maxN;
    float* h_item = h_user + (size_t)NU * DHID;
    float* outF = (float*)d_out;

    const size_t zn = (size_t)maxN * DHID + maxN;   // sum + cnt (contiguous)

    auto run_conv = [&](const float* xsrc, const float* xd, const int* eidx, int E,
                        const float* Wl, const float* Wr, const float* b,
                        float* outp, int relu, int nrows_dst,
                        const int* limit_ptr, int limit_const, const int* off_ptr) {
        zero_kernel<<<(unsigned)((zn + 255) / 256), 256, 0, stream>>>(sumb, zn);
        unsigned ablocks = (unsigned)(((size_t)E * 32 + 255) / 256);
        agg_kernel<<<ablocks, 256, 0, stream>>>(xsrc, eidx, eidx + E, sumb, cntb, E);
        unsigned sblocks = (unsigned)((nrows_dst + 16 * SAGE_WAVES - 1) / (16 * SAGE_WAVES));
        sage_kernel<<<sblocks, 32 * SAGE_WAVES, 0, stream>>>(
            sumb, cntb, xd, Wl, Wr, b, outp, relu, nrows_dst,
            limit_ptr, limit_const, off_ptr);
    };

    // Layer 1 (normalize then ReLU), full node sets
    run_conv(x_user, x_item, eui, E_ui, l1_ui_Wl, l1_ui_Wr, l1_ui_b,
             h_item, 1, NI, nullptr, NI, nullptr);
    run_conv(x_item, x_user, eiu, E_iu, l1_iu_Wl, l1_iu_Wr, l1_iu_b,
             h_user, 1, NU, nullptr, NU, nullptr);

    // Layer 2: write seed rows straight into d_out = [o_user[:B] | o_item[:B]]
    // o_item half starts at batch_size_user * 128 (device scalar -> off_ptr)
    run_conv(h_user, h_item, eui, E_ui, l2_ui_Wl, l2_ui_Wr, l2_ui_b,
             outF, 0, NI, bsi, NI, bsu);
    run_conv(h_item, h_user, eiu, E_iu, l2_iu_Wl, l2_iu_Wr, l2_iu_b,
             outF, 0, NU, bsu, NU, nullptr);
}